// ReformerAttention_27247272526198
// MI455X (gfx1250) — compile-verified
//
#include <hip/hip_runtime.h>
#include <math.h>

#define BATCH 8
#define SEQ   8192
#define DIM   64
#define NHASH 8
#define NBUCK 128                 // buckets per hash round
#define NBALL (NHASH*NBUCK)       // 1024 total buckets
#define BSZ   64                  // bucket/chunk size
#define NCHUNK (NHASH*(SEQ/BSZ))  // 1024 chunks per batch
#define HS    (NHASH*SEQ)         // 65536
#define TS    512                 // sort tile size
#define NT    (SEQ/TS)            // 16 tiles per (b,h)
#define SELF_VAL (-5.0e4f)

typedef float v2f __attribute__((ext_vector_type(2)));
typedef float v8f __attribute__((ext_vector_type(8)));

// ---------------------------------------------------------------------------
// Kernel 1: LSH hash.  rotated = qk @ rot[:,h,:]  (64x64)·(64x64) via WMMA,
// bucket = argmax over [rotated, -rotated] (first-max tie-break), histogram.
// ---------------------------------------------------------------------------
__global__ __launch_bounds__(256) void lsh_hash_kernel(
    const float* __restrict__ qk, const float* __restrict__ rot,
    int* __restrict__ bucketArr, int* __restrict__ tilecnt)
{
  __shared__ __align__(16) float A [64][68];
  __shared__ __align__(16) float Bm[64][68];
  __shared__ __align__(16) float R [64][68];

  const int bid = blockIdx.x;
  const int tt  = bid & 127;          // 128 row-tiles of 64
  const int h   = (bid >> 7) & 7;
  const int b   = bid >> 10;
  const int tid = threadIdx.x;
  const int t0  = tt * 64;

  { // load 64x64 qk tile
    int row = tid >> 2, c0 = (tid & 3) * 16;
    const float4* s4 = (const float4*)(qk + ((size_t)b*SEQ + t0 + row)*DIM + c0);
    float4* d4 = (float4*)(&A[row][c0]);
    d4[0]=s4[0]; d4[1]=s4[1]; d4[2]=s4[2]; d4[3]=s4[3];
  }
  { // load rotations slice for hash h: Bm[f][i] = rot[f, h, i]
    int f = tid >> 2, c0 = (tid & 3) * 16;
    const float4* s4 = (const float4*)(rot + (size_t)f*(NHASH*64) + (size_t)h*64 + c0);
    float4* d4 = (float4*)(&Bm[f][c0]);
    d4[0]=s4[0]; d4[1]=s4[1]; d4[2]=s4[2]; d4[3]=s4[3];
  }
  __syncthreads();

  const int wave = tid >> 5, lane = tid & 31, half = lane >> 4, mn = lane & 15;
  for (int T = wave; T < 16; T += 8) {
    const int mi = T >> 2, ni = T & 3;
    v8f acc = {};
#pragma unroll
    for (int k0 = 0; k0 < 64; k0 += 4) {
      v2f a, bb;
      a.x  = A [mi*16 + mn][k0 + 2*half    ];
      a.y  = A [mi*16 + mn][k0 + 2*half + 1];
      bb.x = Bm[k0 + 2*half    ][ni*16 + mn];
      bb.y = Bm[k0 + 2*half + 1][ni*16 + mn];
      acc = __builtin_amdgcn_wmma_f32_16x16x4_f32(false, a, false, bb,
                                                  (short)0, acc, false, false);
    }
#pragma unroll
    for (int r = 0; r < 8; r++)
      R[mi*16 + r + 8*half][ni*16 + mn] = acc[r];
  }
  __syncthreads();

  if (tid < 64) {
    const int row = tid;
    float best = R[row][0]; int bidx = 0;
    for (int j = 1; j < 128; j++) {
      float vv = (j < 64) ? R[row][j] : -R[row][j - 64];
      if (vv > best) { best = vv; bidx = j; }
    }
    const int q = h*NBUCK + bidx;       // global bucket (offsets folded in)
    const int t = t0 + row;
    bucketArr[((size_t)b*NHASH + h)*SEQ + t] = q;
    atomicAdd(&tilecnt[((size_t)b*NBALL + q)*NT + (t >> 9)], 1);
  }
}

// ---------------------------------------------------------------------------
// Kernel 2: per-batch exclusive scan over 1024 buckets + in-place exclusive
// tile prefixes (global base folded in) -> stable counting-sort bases.
// ---------------------------------------------------------------------------
__global__ __launch_bounds__(1024) void lsh_prefix_kernel(int* __restrict__ tilecnt)
{
  __shared__ int sc[NBALL];
  const int b = blockIdx.x, q = threadIdx.x;
  int* tc = tilecnt + ((size_t)b*NBALL + q)*NT;
  int s = 0;
#pragma unroll
  for (int j = 0; j < NT; j++) s += tc[j];
  sc[q] = s;
  __syncthreads();
  for (int off = 1; off < NBALL; off <<= 1) {
    int vv = (q >= off) ? sc[q - off] : 0;
    __syncthreads();
    sc[q] += vv;
    __syncthreads();
  }
  int run = sc[q] - s;                  // exclusive over buckets
#pragma unroll
  for (int j = 0; j < NT; j++) { int c0 = tc[j]; tc[j] = run; run += c0; }
}

// ---------------------------------------------------------------------------
// Kernel 3: stable rank within (bucket, tile), scatter the sort permutation:
// sticker[b][pos] = h*SEQ + t.
// ---------------------------------------------------------------------------
__global__ __launch_bounds__(TS) void lsh_rank_kernel(
    const int* __restrict__ bucketArr, const int* __restrict__ tilecnt,
    int* __restrict__ sticker)
{
  __shared__ int bk[TS];
  const int bid  = blockIdx.x;
  const int tile = bid & (NT - 1);
  const int h    = (bid >> 4) & 7;
  const int b    = bid >> 7;
  const int tid  = threadIdx.x;
  const int t    = tile*TS + tid;
  const int q    = bucketArr[((size_t)b*NHASH + h)*SEQ + t];
  bk[tid] = q;
  __syncthreads();
  int r = 0;
  for (int j = 0; j < tid; j++) r += (bk[j] == q) ? 1 : 0;
  const int pos = tilecnt[((size_t)b*NBALL + q)*NT + tile] + r;
  sticker[(size_t)b*HS + pos] = h*SEQ + t;
}

// ---------------------------------------------------------------------------
// Kernel 4: bucketed attention chunk. One block per (b, chunk).
// QK^T and P·V via f32 WMMA; scatter output through the sort permutation.
// ---------------------------------------------------------------------------
__global__ __launch_bounds__(256) void lsh_attn_kernel(
    const float* __restrict__ qk, const float* __restrict__ vin,
    const int* __restrict__ sticker,
    float* __restrict__ oBuf, float* __restrict__ lgBuf)
{
  __shared__ int idxA[128];
  __shared__ int tA  [128];
  __shared__ __align__(16) float Qs[64][68];
  __shared__ __align__(16) float Ks[128][68];
  __shared__ __align__(16) float Vs[128][68];
  __shared__ __align__(16) float Ps[64][132];
  __shared__ float lse[64];

  const int bid = blockIdx.x;
  const int c   = bid & (NCHUNK - 1);
  const int b   = bid >> 10;
  const int tid = threadIdx.x;
  const int prev = (c + NCHUNK - 1) & (NCHUNK - 1);   // look-one-back with wrap

  if (tid < 128) {
    const int src_chunk = (tid < 64) ? c : prev;
    const int j = tid & 63;
    const int idx = sticker[(size_t)b*HS + (size_t)src_chunk*BSZ + j];
    idxA[tid] = idx;
    tA[tid]   = idx & (SEQ - 1);        // original token position (st)
  }
  __syncthreads();

  { // gather Q rows (current chunk)
    int row = tid >> 2, c0 = (tid & 3) * 16;
    const float4* s4 = (const float4*)(qk + ((size_t)b*SEQ + tA[row])*DIM + c0);
    float4* d4 = (float4*)(&Qs[row][c0]);
    d4[0]=s4[0]; d4[1]=s4[1]; d4[2]=s4[2]; d4[3]=s4[3];
  }
  { // gather K (raw qk) and V rows (128 kv)
    int row = tid >> 1, c0 = (tid & 1) * 32;
    const int t = tA[row];
    const float4* sk = (const float4*)(qk  + ((size_t)b*SEQ + t)*DIM + c0);
    const float4* sv = (const float4*)(vin + ((size_t)b*SEQ + t)*DIM + c0);
    float4* dk = (float4*)(&Ks[row][c0]);
    float4* dv = (float4*)(&Vs[row][c0]);
#pragma unroll
    for (int i = 0; i < 8; i++) { dk[i] = sk[i]; dv[i] = sv[i]; }
  }
  __syncthreads();

  if (tid < 128) {  // unit-normalize keys: k / (||k|| + 1e-6)
    float ss = 0.f;
#pragma unroll
    for (int d = 0; d < 64; d++) { float x = Ks[tid][d]; ss += x * x; }
    const float inv = 1.0f / (sqrtf(ss) + 1e-6f);
#pragma unroll
    for (int d = 0; d < 64; d++) Ks[tid][d] *= inv;
  }
  __syncthreads();

  const int wave = tid >> 5, lane = tid & 31, half = lane >> 4, mn = lane & 15;

  { // dots = Q · K^T * dim^-0.5, 64x128; wave owns 16-col strip ni=wave
    const int ni = wave;
    for (int mi = 0; mi < 4; mi++) {
      v8f acc = {};
#pragma unroll
      for (int k0 = 0; k0 < 64; k0 += 4) {
        v2f a, bb;
        a.x  = Qs[mi*16 + mn][k0 + 2*half    ];
        a.y  = Qs[mi*16 + mn][k0 + 2*half + 1];
        bb.x = Ks[ni*16 + mn][k0 + 2*half    ];   // B[k][n] = K[n][k]
        bb.y = Ks[ni*16 + mn][k0 + 2*half + 1];
        acc = __builtin_amdgcn_wmma_f32_16x16x4_f32(false, a, false, bb,
                                                    (short)0, acc, false, false);
      }
#pragma unroll
      for (int r = 0; r < 8; r++) {
        const int i = mi*16 + r + 8*half, j = ni*16 + mn;
        float dval = acc[r] * 0.125f;             // 1/sqrt(64)
        if (tA[i] == tA[j]) dval = SELF_VAL;      // self-attention mask
        Ps[i][j] = dval;
      }
    }
  }
  __syncthreads();

  if (tid < 64) {   // row softmax + logsumexp over 128 kv
    float m = -1e30f;
#pragma unroll
    for (int j = 0; j < 128; j++) m = fmaxf(m, Ps[tid][j]);
    float ssum = 0.f;
#pragma unroll
    for (int j = 0; j < 128; j++) { float e = expf(Ps[tid][j] - m); Ps[tid][j] = e; ssum += e; }
    lse[tid] = m + logf(ssum);
    const float invs = 1.0f / ssum;
#pragma unroll
    for (int j = 0; j < 128; j++) Ps[tid][j] *= invs;
  }
  __syncthreads();

  // bo = P (64x128) · V (128x64); 16 tiles, 2 per wave; stage into Qs (dead)
  for (int T = wave; T < 16; T += 8) {
    const int mi = T >> 2, ni = T & 3;
    v8f acc = {};
#pragma unroll
    for (int k0 = 0; k0 < 128; k0 += 4) {
      v2f a, bb;
      a.x  = Ps[mi*16 + mn][k0 + 2*half    ];
      a.y  = Ps[mi*16 + mn][k0 + 2*half + 1];
      bb.x = Vs[k0 + 2*half    ][ni*16 + mn];
      bb.y = Vs[k0 + 2*half + 1][ni*16 + mn];
      acc = __builtin_amdgcn_wmma_f32_16x16x4_f32(false, a, false, bb,
                                                  (short)0, acc, false, false);
    }
#pragma unroll
    for (int r = 0; r < 8; r++)
      Qs[mi*16 + r + 8*half][ni*16 + mn] = acc[r];
  }
  __syncthreads();

  { // scatter rows through undo_sort (direct scatter to original flat index)
    int row = tid >> 2, c0 = (tid & 3) * 16;
    float4* d4 = (float4*)(oBuf + ((size_t)b*HS + idxA[row])*DIM + c0);
    const float4* s4 = (const float4*)(&Qs[row][c0]);
    d4[0]=s4[0]; d4[1]=s4[1]; d4[2]=s4[2]; d4[3]=s4[3];
  }
  if (tid < 64) lgBuf[(size_t)b*HS + idxA[tid]] = lse[tid];
}

// ---------------------------------------------------------------------------
// Kernel 5: combine across hash rounds with softmax(logsumexp) weights.
// ---------------------------------------------------------------------------
__global__ __launch_bounds__(256) void lsh_combine_kernel(
    const float* __restrict__ oBuf, const float* __restrict__ lgBuf,
    float* __restrict__ out)
{
  const int g = blockIdx.x * blockDim.x + threadIdx.x;  // over B*SEQ
  const int b = g >> 13;
  const int t = g & (SEQ - 1);
  float lg[NHASH];
  float m = -1e30f;
#pragma unroll
  for (int h = 0; h < NHASH; h++) {
    lg[h] = lgBuf[(size_t)b*HS + (size_t)h*SEQ + t];
    m = fmaxf(m, lg[h]);
  }
  float ssum = 0.f;
#pragma unroll
  for (int h = 0; h < NHASH; h++) { lg[h] = expf(lg[h] - m); ssum += lg[h]; }
  const float inv = 1.0f / ssum;
#pragma unroll
  for (int h = 0; h < NHASH; h++) lg[h] *= inv;

  float* dst = out + ((size_t)b*SEQ + t)*DIM;
#pragma unroll
  for (int d = 0; d < DIM; d += 4) {
    float4 acc = {0.f, 0.f, 0.f, 0.f};
    for (int h = 0; h < NHASH; h++) {
      const float4 o4 = *(const float4*)(oBuf + ((size_t)b*HS + (size_t)h*SEQ + t)*DIM + d);
      acc.x += lg[h]*o4.x; acc.y += lg[h]*o4.y; acc.z += lg[h]*o4.z; acc.w += lg[h]*o4.w;
    }
    *(float4*)(dst + d) = acc;
  }
}

// ---------------------------------------------------------------------------
extern "C" void kernel_launch(void* const* d_in, const int* in_sizes, int n_in,
                              void* d_out, int out_size, void* d_ws, size_t ws_size,
                              hipStream_t stream)
{
  const float* qk  = (const float*)d_in[0];
  const float* v   = (const float*)d_in[1];
  const float* rot = (const float*)d_in[2];
  // d_in[3] = query_len (== SEQ), compile-time constant here.

  char* ws = (char*)d_ws;
  size_t off = 0;
  auto alloc = [&](size_t bytes) -> void* {
    void* p = ws + off;
    off += (bytes + 255) & ~(size_t)255;
    return p;
  };
  int*   bucketArr = (int*)  alloc((size_t)BATCH*NHASH*SEQ*sizeof(int));
  int*   tilecnt   = (int*)  alloc((size_t)BATCH*NBALL*NT*sizeof(int));
  int*   sticker   = (int*)  alloc((size_t)BATCH*HS*sizeof(int));
  float* oBuf      = (float*)alloc((size_t)BATCH*HS*DIM*sizeof(float));
  float* lgBuf     = (float*)alloc((size_t)BATCH*HS*sizeof(float));
  if (off > ws_size) return;   // workspace too small; nothing safe to do

  hipMemsetAsync(tilecnt, 0, (size_t)BATCH*NBALL*NT*sizeof(int), stream);

  lsh_hash_kernel   <<<BATCH*NHASH*(SEQ/64), 256, 0, stream>>>(qk, rot, bucketArr, tilecnt);
  lsh_prefix_kernel <<<BATCH,               1024, 0, stream>>>(tilecnt);
  lsh_rank_kernel   <<<BATCH*NHASH*NT,        TS, 0, stream>>>(bucketArr, tilecnt, sticker);
  lsh_attn_kernel   <<<BATCH*NCHUNK,         256, 0, stream>>>(qk, v, sticker, oBuf, lgBuf);
  lsh_combine_kernel<<<(BATCH*SEQ)/256,      256, 0, stream>>>(oBuf, lgBuf, (float*)d_out);
}